// BitDense_61289183314095
// MI455X (gfx1250) — compile-verified
//
#include <hip/hip_runtime.h>
#include <stdint.h>

// ---------------------------------------------------------------------------
// BitDense (XNOR-popcount binary dense layer), MI455X / gfx1250.
//
// inputs: uint32[1024]           (32768 input bits, packed)
// w     : uint32[32768][1024]    (128 MB -> HBM-bandwidth bound, ~5.5us floor)
// b     : int32 [32768]
// out   : uint32[1024]           (sign bits of 32768 units, packed)
//
// Strategy: stream weights via CDNA5 async global->LDS DMA (ASYNCcnt),
// double-buffered 256-row x 32-word tiles, one unit per thread, xor+popcount
// in VALU. Column dimension split 4 ways across blocks (partials in d_ws)
// so 512 workgroups keep every WGP busy regardless of WGP count.
// ---------------------------------------------------------------------------

#define UNITS            32768
#define PACKED           1024
#define SPLITS           4
#define BLK_UNITS        256
#define COLS_PER_BLOCK   (PACKED / SPLITS)          // 256 words
#define TILE_COLS        32                          // words per tile
#define NTILES           (COLS_PER_BLOCK / TILE_COLS)// 8
#define ROW_PITCH        36                          // dwords: 32 + 4 pad (bank-conflict-free)
#define TILE_UINTS       (BLK_UNITS * ROW_PITCH)     // 9216 dwords = 36 KB
#define CHUNKS_PER_ROW   (TILE_COLS / 4)             // 8 x 16B chunks per row
#define CHUNKS_PER_TILE  (BLK_UNITS * CHUNKS_PER_ROW)// 2048
#define CHUNKS_PER_THREAD (CHUNKS_PER_TILE / 256)    // 8 async b128 per thread per tile
#define LDS_UINTS        (COLS_PER_BLOCK + 2 * TILE_UINTS)  // 18688 dwords = 74,752 B

#if defined(__has_builtin)
#  if __has_builtin(__builtin_amdgcn_global_load_async_to_lds_b128)
#    define HAVE_ASYNC_BUILTIN 1
#  endif
#  if __has_builtin(__builtin_amdgcn_s_wait_asynccnt)
#    define HAVE_WAIT_BUILTIN 1
#  endif
#endif

// Exact pointee type the builtin expects: 16-byte GCC-style int vector,
// in addrspace(1) (printed by clang as "__device__") / addrspace(3) (LDS).
typedef int v4i_t __attribute__((vector_size(4 * sizeof(int))));
typedef __attribute__((address_space(1))) v4i_t* gvec_ptr;
typedef __attribute__((address_space(3))) v4i_t* lvec_ptr;

__device__ __forceinline__ void async_copy16(const unsigned* g, unsigned* l) {
#if defined(HAVE_ASYNC_BUILTIN)
  __builtin_amdgcn_global_load_async_to_lds_b128(
      (gvec_ptr)(uintptr_t)g,   // global source (inttoptr, drops const)
      (lvec_ptr)l,              // LDS destination (generic -> AS3 cast)
      /*offset=*/0, /*cpol=*/0);
#else
  unsigned lds_off = (unsigned)(uintptr_t)(__attribute__((address_space(3))) void*)l;
  asm volatile("global_load_async_to_lds_b128 %0, %1, off"
               :: "v"(lds_off), "v"(g) : "memory");
#endif
}

__device__ __forceinline__ void wait_async_le8() {
#if defined(HAVE_WAIT_BUILTIN)
  __builtin_amdgcn_s_wait_asynccnt(CHUNKS_PER_THREAD);
#else
  asm volatile("s_wait_asynccnt %0" :: "n"(CHUNKS_PER_THREAD));
#endif
}
__device__ __forceinline__ void wait_async_le0() {
#if defined(HAVE_WAIT_BUILTIN)
  __builtin_amdgcn_s_wait_asynccnt(0);
#else
  asm volatile("s_wait_asynccnt 0");
#endif
}

// Issue one 256-row x 32-word tile (36 KB) of weights as async b128 copies.
// Thread `tid` copies chunks c = i*256+tid; consecutive lanes -> consecutive
// 16B global addresses (coalesced streaming of whole rows).
__device__ __forceinline__ void issue_tile(const unsigned* __restrict__ w,
                                           int base_u, int colw,
                                           unsigned* buf, int tid) {
#pragma unroll
  for (int i = 0; i < CHUNKS_PER_THREAD; ++i) {
    const int c   = i * 256 + tid;
    const int row = c >> 3;      // 8 chunks per row
    const int j   = c & 7;
    const unsigned* g = w + (size_t)(base_u + row) * PACKED + (colw + j * 4);
    unsigned*       l = buf + row * ROW_PITCH + j * 4;
    async_copy16(g, l);
  }
}

__global__ __launch_bounds__(256)
void bitdense_partial_kernel(const unsigned* __restrict__ inputs,
                             const unsigned* __restrict__ w,
                             int* __restrict__ acc) {
  extern __shared__ unsigned smem[];
  unsigned* lds_in = smem;                          // 256 dwords (input slice)
  unsigned* buf0   = smem + COLS_PER_BLOCK;         // tile buffer A
  unsigned* buf1   = buf0 + TILE_UINTS;             // tile buffer B

  const int tid     = (int)threadIdx.x;
  const int ublock  = (int)(blockIdx.x >> 2);       // 0..127 : which 256 units
  const int s       = (int)(blockIdx.x & 3);        // 0..3   : column split
  const int base_u  = ublock * BLK_UNITS;
  const int colbase = s * COLS_PER_BLOCK;

  // Stage this block's 256-word input slice into LDS (1 KB).
  if (tid < COLS_PER_BLOCK / 4) {
    const uint4 v = *(const uint4*)(inputs + colbase + tid * 4);
    *(uint4*)(lds_in + tid * 4) = v;
  }

  // Prologue: prefetch tile 0.
  issue_tile(w, base_u, colbase, buf0, tid);

  unsigned ones = 0;
  int p = 0;
#pragma unroll 1
  for (int t = 0; t < NTILES; ++t) {
    unsigned* cur = p ? buf1 : buf0;
    unsigned* nxt = p ? buf0 : buf1;
    if (t + 1 < NTILES) {
      issue_tile(w, base_u, colbase + (t + 1) * TILE_COLS, nxt, tid);
      wait_async_le8();   // async loads complete in order -> tile t done
    } else {
      wait_async_le0();
    }
    __syncthreads();      // tile t visible to all waves (also covers lds_in at t==0)

    const unsigned* myrow = cur + tid * ROW_PITCH;          // this thread's unit row
    const unsigned* inw   = lds_in + t * TILE_COLS;         // broadcast input chunk
#pragma unroll
    for (int j = 0; j < CHUNKS_PER_ROW; ++j) {
      const uint4 wv = *(const uint4*)(myrow + j * 4);      // ds_load_b128, conflict-free
      const uint4 iv = *(const uint4*)(inw + j * 4);        // uniform addr -> broadcast
      ones += (unsigned)__popc(wv.x ^ iv.x);
      ones += (unsigned)__popc(wv.y ^ iv.y);
      ones += (unsigned)__popc(wv.z ^ iv.z);
      ones += (unsigned)__popc(wv.w ^ iv.w);
    }
    __syncthreads();      // all waves done reading `cur`; safe to refill later
    p ^= 1;
  }

  acc[s * UNITS + base_u + tid] = (int)ones;
}

// Combine the 4 column-split partials, apply bias, binarize, and pack sign
// bits with numpy packbits order: within an output word, bit p <- unit (p^7)
// of that word's 32-unit group (MSB-first per byte, bytes little-endian).
__global__ __launch_bounds__(256)
void bitdense_pack_kernel(const int* __restrict__ acc,
                          const int* __restrict__ bias,
                          unsigned* __restrict__ out) {
  __shared__ unsigned char bools[256];
  const int tid = (int)threadIdx.x;
  const int u   = (int)blockIdx.x * 256 + tid;

  const int ones = acc[u] + acc[UNITS + u] + acc[2 * UNITS + u] + acc[3 * UNITS + u];
  const int val  = 32768 - 2 * ones + bias[u];   // BITS*PACKED_INPUTS = 32768
  bools[tid] = (val < 0) ? 1u : 0u;
  __syncthreads();

  if (tid < 8) {
    unsigned word = 0;
#pragma unroll
    for (int pbit = 0; pbit < 32; ++pbit)
      word |= ((unsigned)bools[tid * 32 + (pbit ^ 7)]) << pbit;
    out[(int)blockIdx.x * 8 + tid] = word;
  }
}

extern "C" void kernel_launch(void* const* d_in, const int* in_sizes, int n_in,
                              void* d_out, int out_size, void* d_ws, size_t ws_size,
                              hipStream_t stream) {
  (void)in_sizes; (void)n_in; (void)out_size; (void)ws_size;
  const unsigned* inputs = (const unsigned*)d_in[0];
  const unsigned* w      = (const unsigned*)d_in[1];
  const int*      bias   = (const int*)d_in[2];
  unsigned*       out    = (unsigned*)d_out;
  int*            acc    = (int*)d_ws;   // SPLITS * UNITS int32 = 512 KB scratch

  const size_t shmem = (size_t)LDS_UINTS * sizeof(unsigned);  // 74,752 B (< 320 KB/WGP)

  bitdense_partial_kernel<<<dim3((UNITS / BLK_UNITS) * SPLITS), dim3(256), shmem, stream>>>(
      inputs, w, acc);
  bitdense_pack_kernel<<<dim3(UNITS / 256), dim3(256), 0, stream>>>(acc, bias, out);
}